// MolecularGNN_81406810128840
// MI455X (gfx1250) — compile-verified
//
#include <hip/hip_runtime.h>

// ---------------------------------------------------------------------------
// MolecularGNN for MI455X (gfx1250, wave32).
// GEMMs use V_WMMA_F32_16X16X4_F32 (exact fp32 on the matrix pipe).
// Edge gather/scatter: one wave per edge, float2 per lane, f32 global atomics
// (working set ~77MB lives in the 192MB L2, so atomics/gathers hit L2).
// ---------------------------------------------------------------------------

typedef float v2f __attribute__((ext_vector_type(2)));
typedef float v8f __attribute__((ext_vector_type(8)));

__device__ __forceinline__ v8f wmma4(v2f a, v2f b, v8f c) {
  // 8 args: (neg_a, A, neg_b, B, c_mod, C, reuse_a, reuse_b)
  return __builtin_amdgcn_wmma_f32_16x16x4_f32(false, a, false, b, (short)0, c,
                                               false, false);
}

__global__ void k_zero(float* __restrict__ p, int n) {
  int i = blockIdx.x * blockDim.x + threadIdx.x;
  if (i < n) p[i] = 0.0f;
}

__global__ void k_deg(const int* __restrict__ dst, float* __restrict__ deg, int E) {
  int e = blockIdx.x * blockDim.x + threadIdx.x;
  if (e < E) atomicAdd(&deg[dst[e]], 1.0f);
}

__global__ void k_dinv(float* __restrict__ dinv, int N) {
  int i = blockIdx.x * blockDim.x + threadIdx.x;
  if (i < N) dinv[i] = rsqrtf(dinv[i] + 1.0f);
}

// C[n,64] = A[n,K] @ B[K,64] (+bias).  One wave -> 16x64 tile via 4x v8f acc.
// A-frag (f32 16x4): lane<16 holds K=k,k+1 ; lane>=16 holds K=k+2,k+3, M=lane&15.
// B-frag (f32 4x16): same K split across lane halves, N = lane&15.
__global__ void __launch_bounds__(256) k_gemm(const float* __restrict__ A,
                                              const float* __restrict__ B,
                                              const float* __restrict__ bias,
                                              float* __restrict__ C,
                                              int nRows, int K) {
  __shared__ float Bs[64 * 64];  // up to 16KB (LDS per WGP is 320KB)
  int t = threadIdx.x;
  for (int i = t; i < K * 64; i += blockDim.x) Bs[i] = B[i];
  __syncthreads();

  int lane = t & 31;
  int wave = t >> 5;
  int m0 = (blockIdx.x * (blockDim.x >> 5) + wave) * 16;
  if (m0 >= nRows) return;  // wave-uniform: EXEC stays all-ones for WMMA

  const int mrow  = lane & 15;
  const int khalf = (lane >> 4) << 1;  // 0 or 2

  v8f acc[4] = {};
  for (int k = 0; k < K; k += 4) {
    const float* ap = A + (m0 + mrow) * K + (k + khalf);
    v2f a;
    a.x = ap[0];
    a.y = ap[1];
#pragma unroll
    for (int nt = 0; nt < 4; ++nt) {
      v2f b;
      b.x = Bs[(k + khalf) * 64 + nt * 16 + mrow];
      b.y = Bs[(k + khalf + 1) * 64 + nt * 16 + mrow];
      acc[nt] = wmma4(a, b, acc[nt]);
    }
  }

  // D layout: VGPR r -> row m0 + r + 8*(lane>>4), col = nt*16 + (lane&15)
  int rbase = m0 + ((lane >> 4) << 3);
#pragma unroll
  for (int nt = 0; nt < 4; ++nt) {
    int col = nt * 16 + mrow;
    float bv = bias ? bias[col] : 0.0f;
#pragma unroll
    for (int r = 0; r < 8; ++r)
      C[(rbase + r) * 64 + col] = acc[nt][r] + bv;
  }
}

// agg = hw * dinv^2   (self-loop term; also fully initializes agg)
__global__ void k_selfloop(const float* __restrict__ hw,
                           const float* __restrict__ dinv,
                           float* __restrict__ agg, int total) {
  int i = blockIdx.x * blockDim.x + threadIdx.x;
  if (i >= total) return;
  int n = i >> 6;
  float di = dinv[n];
  agg[i] = hw[i] * di * di;
}

// One wave per edge: 32 lanes x float2 = 64 channels.  Gathers + atomics hit L2.
__global__ void k_scatter(const float* __restrict__ hw,
                          const float* __restrict__ dinv,
                          const int* __restrict__ src,
                          const int* __restrict__ dst,
                          float* __restrict__ agg, int E) {
  int tid = blockIdx.x * blockDim.x + threadIdx.x;
  int e = tid >> 5;
  if (e >= E) return;
  int lane = tid & 31;
  int s = src[e];
  int d = dst[e];
  float sc = dinv[s] * dinv[d];
  const float2 v = *(const float2*)(hw + s * 64 + lane * 2);
  atomicAdd(&agg[d * 64 + lane * 2 + 0], v.x * sc);
  atomicAdd(&agg[d * 64 + lane * 2 + 1], v.y * sc);
}

// Per-channel sum / sumsq of (agg + conv_b) -> stats[0..63]=sum, [64..127]=sumsq
__global__ void k_bnstats(const float* __restrict__ agg,
                          const float* __restrict__ bvec,
                          float* __restrict__ stats, int N) {
  int t = threadIdx.x;            // 256 threads: 4 row-lanes x 64 channels
  int c = t & 63;
  int r0 = blockIdx.x * 4 + (t >> 6);
  int stride = gridDim.x * 4;
  float bc = bvec[c];
  float s = 0.0f, q = 0.0f;
  for (int r = r0; r < N; r += stride) {
    float v = agg[r * 64 + c] + bc;
    s += v;
    q += v * v;
  }
  atomicAdd(&stats[c], s);
  atomicAdd(&stats[64 + c], q);
}

__global__ void k_bnfinal(const float* __restrict__ stats,
                          const float* __restrict__ gamma,
                          const float* __restrict__ beta,
                          float* __restrict__ sscale, float* __restrict__ sshift,
                          float invN) {
  int c = threadIdx.x;  // 64
  float mean = stats[c] * invN;
  float var = stats[64 + c] * invN - mean * mean;
  float sc = gamma[c] * rsqrtf(var + 1e-5f);
  sscale[c] = sc;
  sshift[c] = beta[c] - mean * sc;
}

// h = relu( (agg + b) * scale + shift ) + h     (fused BN + ReLU + residual)
__global__ void k_apply(const float* __restrict__ agg,
                        const float* __restrict__ bvec,
                        const float* __restrict__ sscale,
                        const float* __restrict__ sshift,
                        float* __restrict__ h, int total) {
  int i = blockIdx.x * blockDim.x + threadIdx.x;
  if (i >= total) return;
  int c = i & 63;
  float v = (agg[i] + bvec[c]) * sscale[c] + sshift[c];
  h[i] = fmaxf(v, 0.0f) + h[i];
}

// One wave per node: accumulate row into pooled sums + counts.
__global__ void k_pool(const float* __restrict__ h, const int* __restrict__ batch,
                       float* __restrict__ psum, float* __restrict__ pcnt, int N) {
  int tid = blockIdx.x * blockDim.x + threadIdx.x;
  int n = tid >> 5;
  if (n >= N) return;
  int lane = tid & 31;
  int g = batch[n];
  const float2 v = *(const float2*)(h + n * 64 + lane * 2);
  atomicAdd(&psum[g * 64 + lane * 2 + 0], v.x);
  atomicAdd(&psum[g * 64 + lane * 2 + 1], v.y);
  if (lane == 0) atomicAdd(&pcnt[g], 1.0f);
}

// out[g] = relu(pooled @ w1 + b1) @ w2 + b2  (tiny: 256 x 64 x 32)
__global__ void k_mlp(const float* __restrict__ psum, const float* __restrict__ pcnt,
                      const float* __restrict__ w1, const float* __restrict__ b1,
                      const float* __restrict__ w2, const float* __restrict__ b2,
                      float* __restrict__ out, int G) {
  int g = blockIdx.x * blockDim.x + threadIdx.x;
  if (g >= G) return;
  float inv = 1.0f / fmaxf(pcnt[g], 1.0f);
  float pooled[64];
#pragma unroll
  for (int c = 0; c < 64; ++c) pooled[c] = psum[g * 64 + c] * inv;
  float o = 0.0f;
  for (int j = 0; j < 32; ++j) {
    float a = b1[j];
#pragma unroll
    for (int c = 0; c < 64; ++c) a += pooled[c] * w1[c * 32 + j];
    o += fmaxf(a, 0.0f) * w2[j];
  }
  out[g] = o + b2[0];
}

extern "C" void kernel_launch(void* const* d_in, const int* in_sizes, int n_in,
                              void* d_out, int out_size, void* d_ws, size_t ws_size,
                              hipStream_t stream) {
  const float* x        = (const float*)d_in[0];
  const int*   edge     = (const int*)d_in[1];
  const int*   batch    = (const int*)d_in[2];
  const float* w_embed  = (const float*)d_in[3];
  const float* b_embed  = (const float*)d_in[4];
  const float* conv_W   = (const float*)d_in[5];
  const float* conv_b   = (const float*)d_in[6];
  const float* bn_gamma = (const float*)d_in[7];
  const float* bn_beta  = (const float*)d_in[8];
  const float* w1       = (const float*)d_in[9];
  const float* b1       = (const float*)d_in[10];
  const float* w2       = (const float*)d_in[11];
  const float* b2       = (const float*)d_in[12];

  const int N = in_sizes[2];          // 100000
  const int E = in_sizes[1] / 2;      // 1600000
  const int F = in_sizes[0] / N;      // 32
  const int L = in_sizes[6] / 64;     // 4
  const int G = out_size;             // 256
  const int* srcIdx = edge;
  const int* dstIdx = edge + E;

  // workspace carve (floats): 3*N*64 + N + 256 + G*64 + G  (~77.3 MB)
  float* ws     = (float*)d_ws;
  float* h      = ws; ws += (size_t)N * 64;
  float* hw     = ws; ws += (size_t)N * 64;
  float* agg    = ws; ws += (size_t)N * 64;
  float* dinv   = ws; ws += N;
  float* stats  = ws; ws += 128;
  float* sscale = ws; ws += 64;
  float* sshift = ws; ws += 64;
  float* psum   = ws; ws += (size_t)G * 64;
  float* pcnt   = ws; ws += G;

  auto cdiv = [](int a, int b) { return (a + b - 1) / b; };
  dim3 blk(256);

  // degree -> dinv (depends only on edges)
  k_zero<<<cdiv(N, 256), blk, 0, stream>>>(dinv, N);
  k_deg<<<cdiv(E, 256), blk, 0, stream>>>(dstIdx, dinv, E);
  k_dinv<<<cdiv(N, 256), blk, 0, stream>>>(dinv, N);

  const int tiles = cdiv(N, 16);
  const int gemmBlocks = cdiv(tiles, 8);  // 8 waves / block
  const int total = N * 64;

  // node embedding: h = x @ w_embed + b_embed
  k_gemm<<<gemmBlocks, blk, 0, stream>>>(x, w_embed, b_embed, h, N, F);

  for (int l = 0; l < L; ++l) {
    k_gemm<<<gemmBlocks, blk, 0, stream>>>(h, conv_W + l * 64 * 64, nullptr, hw, N, 64);
    k_selfloop<<<cdiv(total, 256), blk, 0, stream>>>(hw, dinv, agg, total);
    k_scatter<<<cdiv(E * 32, 256), blk, 0, stream>>>(hw, dinv, srcIdx, dstIdx, agg, E);
    k_zero<<<1, 128, 0, stream>>>(stats, 128);
    k_bnstats<<<256, blk, 0, stream>>>(agg, conv_b + l * 64, stats, N);
    k_bnfinal<<<1, 64, 0, stream>>>(stats, bn_gamma + l * 64, bn_beta + l * 64,
                                    sscale, sshift, 1.0f / (float)N);
    k_apply<<<cdiv(total, 256), blk, 0, stream>>>(agg, conv_b + l * 64, sscale,
                                                  sshift, h, total);
  }

  // global mean pool + MLP head
  k_zero<<<cdiv(G * 64 + G, 256), blk, 0, stream>>>(psum, G * 64 + G);
  k_pool<<<cdiv(N * 32, 256), blk, 0, stream>>>(h, batch, psum, pcnt, N);
  k_mlp<<<1, G, 0, stream>>>(psum, pcnt, w1, b1, w2, b2, (float*)d_out, G);
}